// KCLLoss_54855322304752
// MI455X (gfx1250) — compile-verified
//
#include <hip/hip_runtime.h>
#include <math.h>
#include <stdint.h>

#define D_   64
#define N_   8192
#define DIM_ 256
#define TOPK_ 32
#define P_   63
#define TEMP_ 0.1f
#define EPS_  1e-8f

#define BLOCKS_PER_D  16
#define ROWS_PER_BLOCK (N_ / BLOCKS_PER_D)   // 512
#define ROWS_PER_WAVE  (ROWS_PER_BLOCK / 8)  // 64

typedef float    f4    __attribute__((ext_vector_type(4)));
typedef float    v2f   __attribute__((ext_vector_type(2)));
typedef float    v8f   __attribute__((ext_vector_type(8)));
typedef unsigned u32x4 __attribute__((ext_vector_type(4)));
typedef int      i32x8 __attribute__((ext_vector_type(8)));

// ---------------------------------------------------------------- k_zero
__global__ __launch_bounds__(256) void k_zero(float* __restrict__ p) {
  p[(size_t)blockIdx.x * 256 + threadIdx.x] = 0.0f;
}

// ---------------------------------------------------------------- k_stream
// One pass over 512 MiB: per-dim sums (atomics) + per-row L2 norms.
__global__ __launch_bounds__(256) void k_stream(const float* __restrict__ emb,
                                                float* __restrict__ sum_all,
                                                float* __restrict__ norms) {
  const int d    = blockIdx.x / BLOCKS_PER_D;
  const int blk  = blockIdx.x % BLOCKS_PER_D;
  const int wave = threadIdx.x >> 5;
  const int lane = threadIdx.x & 31;

  const float* base = emb + (size_t)d * N_ * DIM_;
  const int nStart = blk * ROWS_PER_BLOCK + wave * ROWS_PER_WAVE;

  f4 acc0 = {0.f, 0.f, 0.f, 0.f};   // dims 4*lane   .. 4*lane+3
  f4 acc1 = {0.f, 0.f, 0.f, 0.f};   // dims 128+4*lane ..

  for (int r = 0; r < ROWS_PER_WAVE; ++r) {
    const int n = nStart + r;
    const f4* rowp = (const f4*)(base + (size_t)n * DIM_);
    __builtin_prefetch(rowp + (DIM_ / 4) + lane, 0, 0);  // next row
    f4 v0 = __builtin_nontemporal_load(rowp + lane);
    f4 v1 = __builtin_nontemporal_load(rowp + 32 + lane);
    acc0 += v0;
    acc1 += v1;
    float ss = v0.x*v0.x + v0.y*v0.y + v0.z*v0.z + v0.w*v0.w
             + v1.x*v1.x + v1.y*v1.y + v1.z*v1.z + v1.w*v1.w;
    #pragma unroll
    for (int off = 16; off > 0; off >>= 1)
      ss += __shfl_down(ss, off, 32);
    if (lane == 0) norms[(size_t)d * N_ + n] = sqrtf(ss);
  }

  float* s = sum_all + d * DIM_;
  atomicAdd(&s[4 * lane + 0], acc0.x);
  atomicAdd(&s[4 * lane + 1], acc0.y);
  atomicAdd(&s[4 * lane + 2], acc0.z);
  atomicAdd(&s[4 * lane + 3], acc0.w);
  atomicAdd(&s[128 + 4 * lane + 0], acc1.x);
  atomicAdd(&s[128 + 4 * lane + 1], acc1.y);
  atomicAdd(&s[128 + 4 * lane + 2], acc1.z);
  atomicAdd(&s[128 + 4 * lane + 3], acc1.w);
}

// ---------------------------------------------------------------- k_topk
// Per d: top-32 rows by norm, hard-negative sum, build + normalize samples.
__global__ __launch_bounds__(256) void k_topk(const float* __restrict__ emb,
                                              const float* __restrict__ sum_all,
                                              const float* __restrict__ norms,
                                              float* __restrict__ nS) {
  const int d   = blockIdx.x;
  const int tid = threadIdx.x;

  __shared__ float snorm[N_];      // 32 KB
  __shared__ float red[256];
  __shared__ int   redi[256];
  __shared__ int   sel[TOPK_];
  __shared__ float bc[2];

  for (int n = tid; n < N_; n += 256)
    snorm[n] = norms[(size_t)d * N_ + n];
  __syncthreads();

  for (int k = 0; k < TOPK_; ++k) {
    float bv = -INFINITY; int bi = 0x7fffffff;
    for (int n = tid; n < N_; n += 256) {
      float v = snorm[n];
      if (v > bv) { bv = v; bi = n; }
    }
    red[tid] = bv; redi[tid] = bi;
    __syncthreads();
    for (int s = 128; s > 0; s >>= 1) {
      if (tid < s) {
        float ov = red[tid + s]; int oi = redi[tid + s];
        if (ov > red[tid] || (ov == red[tid] && oi < redi[tid])) {
          red[tid] = ov; redi[tid] = oi;
        }
      }
      __syncthreads();
    }
    if (tid == 0) { sel[k] = redi[0]; snorm[redi[0]] = -INFINITY; }
    __syncthreads();
  }

  // thread tid owns dimension tid (256 == DIM_)
  float ts = 0.f;
  #pragma unroll 4
  for (int k = 0; k < TOPK_; ++k)
    ts += emb[((size_t)d * N_ + sel[k]) * DIM_ + tid];

  const float sa = sum_all[d * DIM_ + tid];
  const float sn = sa - ts;

  // ||sample_all||
  red[tid] = sa * sa;
  __syncthreads();
  for (int s = 128; s > 0; s >>= 1) {
    if (tid < s) red[tid] += red[tid + s];
    __syncthreads();
  }
  if (tid == 0) bc[0] = red[0];
  __syncthreads();

  // ||sample_neg||
  red[tid] = sn * sn;
  __syncthreads();
  for (int s = 128; s > 0; s >>= 1) {
    if (tid < s) red[tid] += red[tid + s];
    __syncthreads();
  }
  if (tid == 0) bc[1] = red[0];
  __syncthreads();

  const float na = fmaxf(sqrtf(bc[0]), EPS_);
  const float nb = fmaxf(sqrtf(bc[1]), EPS_);
  nS[(size_t)d * DIM_ + tid]        = sa / na;
  nS[(size_t)(D_ + d) * DIM_ + tid] = sn / nb;
}

// ---------------------------------------------------------------- k_gram
// G = nS * nS^T  (128x128, K=256) via V_WMMA_F32_16X16X4_F32.
// Staging: each of the 8 waves issues one TDM tensor_load_to_lds for its
// 16x256 f32 slice (16 KB), then waits on TENSORcnt and barriers.
__global__ __launch_bounds__(256) void k_gram(const float* __restrict__ nS,
                                              float* __restrict__ G) {
  extern __shared__ float lds[];                 // 128*256 f32 = 128 KB

  const int wave   = threadIdx.x >> 5;
  const int lane   = threadIdx.x & 31;
  const int lhalf  = lane >> 4;     // 0: lanes 0-15, 1: lanes 16-31
  const int lm     = lane & 15;
  const int m0     = wave * 16;

  // --- TDM descriptor (D#) per ISA 8.3/8.4: 2-D tensor, groups 0+1 only ---
  {
    const unsigned long long ga =
        (unsigned long long)(uintptr_t)nS +
        (unsigned long long)wave * 16ull * DIM_ * 4ull;
    u32x4 g0;
    g0.x = 1u;                                    // count=1 (valid user D#)
    g0.y = (unsigned)(wave * 16 * DIM_ * 4);      // lds_addr (bytes, base 0)
    g0.z = (unsigned)(ga & 0xffffffffu);          // global_addr[31:0]
    g0.w = (unsigned)((ga >> 32) & 0x01ffffffu)   // global_addr[56:32]
         | (2u << 30);                            // type = 2 ("image")
    i32x8 g1;
    g1[0] = 0x20000;                              // data_size=2 (4B), mask=0
    g1[1] = DIM_ << 16;                           // tensor_dim0 = 256
    g1[2] = 16 << 16;                             // tensor_dim1 = 16
    g1[3] = DIM_ << 16;                           // tile_dim0 = 256
    g1[4] = 16;                                   // tile_dim1 = 16
    g1[5] = DIM_;                                 // tensor_dim0_stride = 256
    g1[6] = 0;
    g1[7] = 0;
    asm volatile("tensor_load_to_lds %0, %1" :: "s"(g0), "s"(g1) : "memory");
    __builtin_amdgcn_s_wait_tensorcnt(0);
  }
  __syncthreads();

  for (int c = 0; c < 8; ++c) {
    const int n0 = c * 16;
    v8f acc = {0.f, 0.f, 0.f, 0.f, 0.f, 0.f, 0.f, 0.f};
    for (int kk = 0; kk < DIM_; kk += 4) {
      const int kA = kk + 2 * lhalf;            // lanes 0-15: K,K+1; 16-31: K+2,K+3
      v2f a = *(const v2f*)(&lds[(m0 + lm) * DIM_ + kA]);
      v2f b = *(const v2f*)(&lds[(n0 + lm) * DIM_ + kA]);
      acc = __builtin_amdgcn_wmma_f32_16x16x4_f32(
          /*neg_a=*/false, a, /*neg_b=*/false, b,
          /*c_mod=*/(short)0, acc, /*reuse_a=*/false, /*reuse_b=*/false);
    }
    // C/D layout: VGPR r -> lanes 0-15: (M=r, N=lane); lanes 16-31: (M=r+8, N=lane-16)
    #pragma unroll
    for (int r = 0; r < 8; ++r) {
      const int M = m0 + r + 8 * lhalf;
      const int Nc = n0 + lm;
      G[M * (2 * D_) + Nc] = acc[r];
    }
  }
}

// ---------------------------------------------------------------- k_loss
__global__ __launch_bounds__(64) void k_loss(const float* __restrict__ G,
                                             const int* __restrict__ neg_idx,
                                             float* __restrict__ out) {
  const int p = threadIdx.x;
  __shared__ float red[64];

  float loss = 0.f;
  if (p < P_) {
    const int i = p, j = p + 1;
    const float pos = G[i * (2 * D_) + j];

    float ni[TOPK_], nj[TOPK_];
    #pragma unroll 4
    for (int k = 0; k < TOPK_; ++k) {
      const int idx = neg_idx[p * TOPK_ + k];
      ni[k] = G[i * (2 * D_) + idx];
      nj[k] = G[j * (2 * D_) + idx];
    }

    const float invT = 1.0f / TEMP_;

    float m = pos;
    #pragma unroll
    for (int k = 0; k < TOPK_; ++k) m = fmaxf(m, ni[k]);
    float s = expf((pos - m) * invT);
    #pragma unroll
    for (int k = 0; k < TOPK_; ++k) s += expf((ni[k] - m) * invT);
    loss += (m * invT + logf(s)) - pos * invT;

    m = pos;
    #pragma unroll
    for (int k = 0; k < TOPK_; ++k) m = fmaxf(m, nj[k]);
    s = expf((pos - m) * invT);
    #pragma unroll
    for (int k = 0; k < TOPK_; ++k) s += expf((nj[k] - m) * invT);
    loss += (m * invT + logf(s)) - pos * invT;
  }

  red[p] = loss;
  __syncthreads();
  for (int s = 32; s > 0; s >>= 1) {
    if (p < s) red[p] += red[p + s];
    __syncthreads();
  }
  if (p == 0) out[0] = red[0] / (2.0f * (float)P_);
}

// ---------------------------------------------------------------- launch
extern "C" void kernel_launch(void* const* d_in, const int* in_sizes, int n_in,
                              void* d_out, int out_size, void* d_ws, size_t ws_size,
                              hipStream_t stream) {
  const float* emb = (const float*)d_in[0];       // [64, 8192, 256] f32
  const int*   neg = (const int*)d_in[1];         // [63, 32] i32

  float* ws      = (float*)d_ws;
  float* sum_all = ws;                            // 64*256        = 16384
  float* norms   = sum_all + D_ * DIM_;           // 64*8192       = 524288
  float* nS      = norms + (size_t)D_ * N_;       // 128*256       = 32768
  float* G       = nS + 2 * D_ * DIM_;            // 128*128       = 16384

  // k_gram requests 128 KB dynamic LDS (WGP has 320 KB on CDNA5).
  (void)hipFuncSetAttribute((const void*)k_gram,
                            hipFuncAttributeMaxDynamicSharedMemorySize,
                            2 * D_ * DIM_ * (int)sizeof(float));

  k_zero  <<<D_, 256, 0, stream>>>(sum_all);
  k_stream<<<D_ * BLOCKS_PER_D, 256, 0, stream>>>(emb, sum_all, norms);
  k_topk  <<<D_, 256, 0, stream>>>(emb, sum_all, norms, nS);
  k_gram  <<<1, 256, 2 * D_ * DIM_ * sizeof(float), stream>>>(nS, G);
  k_loss  <<<1, 64, 0, stream>>>(G, neg, (float*)d_out);
}